// SparseDepthWise3D_14310831030995
// MI455X (gfx1250) — compile-verified
//
#include <hip/hip_runtime.h>
#include <hip/hip_bf16.h>
#include <stdint.h>

// ---------------------------------------------------------------------------
// Sparse 3D depthwise conv (3x3x3), N~600K voxels, C=32 channels, f32.
// Pipeline: keys+histogram -> scan -> bucket scatter -> in-bucket rank sort ->
//           wave-per-point conv (lane = channel), 9 lane-parallel searches
//           (dz-triplets share consecutive keys), __shfl broadcast, FMA.
// ---------------------------------------------------------------------------

#define TPB     256
#define NBUCK   16384      // 2^14 buckets over 26-bit keys
#define BSHIFT  12         // bucket = key >> 12
#define MAXBKT  1024       // LDS capacity per bucket in rank kernel
#define PDIM    258        // padded spatial dim (256 + 2*R)

// CDNA5 async global->LDS staging (builtin verified present in round 1)
#if defined(__AMDGCN__) && __has_builtin(__builtin_amdgcn_global_load_async_to_lds_b32)
#define USE_ASYNC_LDS 1
#endif

__device__ __forceinline__ int voxel_key(int b, int x, int y, int z) {
    // padded coords, lexicographic-linear key; max ~51.5M < 2^26
    return (((b + 1) * PDIM + (x + 1)) * PDIM + (y + 1)) * PDIM + (z + 1);
}

__device__ __forceinline__ void copy_f32_to_lds(const float* g, float* l) {
#ifdef USE_ASYNC_LDS
    __builtin_amdgcn_global_load_async_to_lds_b32(
        (__attribute__((address_space(1))) int*)(uintptr_t)g,
        (__attribute__((address_space(3))) int*)(uint32_t)(uintptr_t)l,
        /*offset=*/0, /*cpol=*/0);
#else
    *l = *g;
#endif
}

// ---------------------------------------------------------------------------
__global__ void k_zero(unsigned* p, int n) {
    int i = blockIdx.x * blockDim.x + threadIdx.x;
    if (i < n) p[i] = 0u;
}

__global__ void k_keys_hist(const int* __restrict__ coords,
                            int* __restrict__ keys,
                            unsigned* __restrict__ counts, int n) {
    int i = blockIdx.x * blockDim.x + threadIdx.x;
    if (i >= n) return;
    int b = coords[4 * i + 0];
    int x = coords[4 * i + 1];
    int y = coords[4 * i + 2];
    int z = coords[4 * i + 3];
    int key = voxel_key(b, x, y, z);
    keys[i] = key;
    atomicAdd(&counts[key >> BSHIFT], 1u);
}

// single block, TPB threads, chunked exclusive scan over NBUCK counters
__global__ void k_scan(const unsigned* __restrict__ counts,
                       unsigned* __restrict__ bstart, int nb) {
    __shared__ unsigned s[TPB];
    int tid = threadIdx.x;
    int chunk = nb / TPB;              // 64
    int base = tid * chunk;
    unsigned sum = 0;
    for (int j = 0; j < chunk; ++j) sum += counts[base + j];
    s[tid] = sum;
    __syncthreads();
    for (int off = 1; off < TPB; off <<= 1) {   // Hillis-Steele inclusive
        unsigned v = (tid >= off) ? s[tid - off] : 0u;
        __syncthreads();
        s[tid] += v;
        __syncthreads();
    }
    unsigned run = s[tid] - sum;       // exclusive base for this chunk
    for (int j = 0; j < chunk; ++j) {
        bstart[base + j] = run;
        run += counts[base + j];
    }
    if (tid == TPB - 1) bstart[nb] = run;   // == N
}

__global__ void k_scatter(const int* __restrict__ keys,
                          const unsigned* __restrict__ bstart,
                          unsigned* __restrict__ cursors,
                          int* __restrict__ tmpKey, int* __restrict__ tmpIdx, int n) {
    int i = blockIdx.x * blockDim.x + threadIdx.x;
    if (i >= n) return;
    int key = keys[i];
    int b = key >> BSHIFT;
    unsigned p = bstart[b] + atomicAdd(&cursors[b], 1u);
    tmpKey[p] = key;
    tmpIdx[p] = i;
}

// one block per bucket: deterministic rank sort (keys are unique)
__global__ void k_rank(const int* __restrict__ tmpKey, const int* __restrict__ tmpIdx,
                       const unsigned* __restrict__ bstart,
                       int* __restrict__ skeys, int* __restrict__ order) {
    __shared__ int lk[MAXBKT];
    int b = blockIdx.x;
    int start = (int)bstart[b];
    int m = (int)bstart[b + 1] - start;
    if (m <= 0) return;
    int tid = threadIdx.x;
    if (m <= MAXBKT) {
        for (int e = tid; e < m; e += blockDim.x) lk[e] = tmpKey[start + e];
        __syncthreads();
        for (int e = tid; e < m; e += blockDim.x) {
            int k = lk[e];
            int r = 0;
            for (int j = 0; j < m; ++j) r += (lk[j] < k);
            skeys[start + r] = k;
            order[start + r] = tmpIdx[start + e];
        }
    } else {  // statistically unreachable fallback (avg bucket ~48)
        for (int e = tid; e < m; e += blockDim.x) {
            int k = tmpKey[start + e];
            int r = 0;
            for (int j = 0; j < m; ++j) r += (tmpKey[start + j] < k);
            skeys[start + r] = k;
            order[start + r] = tmpIdx[start + e];
        }
    }
}

// wave-per-point conv; lane == channel (C == 32 == wave32).
// Lanes 0..8 each resolve one (dx,dy) group: the three dz neighbors have
// CONSECUTIVE keys (qbase-1, qbase, qbase+1), so one lower_bound + a <=3-entry
// forward scan yields all three taps. Results broadcast via __shfl.
__global__ void k_conv(const float* __restrict__ feats,
                       const int* __restrict__ coords,
                       const float* __restrict__ weight,
                       const int* __restrict__ skeys,
                       const int* __restrict__ order,
                       float* __restrict__ out,
                       float* __restrict__ out_coords,
                       int n, int C) {
    __shared__ float wlds[32 * 27];
    int tid = threadIdx.x;
    int CW = C * 27;
    for (int t = tid; t < CW; t += blockDim.x)
        copy_f32_to_lds(weight + t, &wlds[t]);     // CDNA5 async -> LDS
#ifdef USE_ASYNC_LDS
    asm volatile("s_wait_asynccnt 0" ::: "memory");
#endif
    __syncthreads();

    int gtid = blockIdx.x * blockDim.x + tid;
    int i    = gtid >> 5;      // sorted point index (one wave each)
    int lane = gtid & 31;
    if (i >= n) return;

    int key = skeys[i];
    int j0  = order[i];
    if (lane < 4)
        out_coords[(size_t)i * 4 + lane] = (float)coords[(size_t)j0 * 4 + lane];

    __builtin_prefetch(&feats[(size_t)j0 * C], 0, 1);  // self row -> global_prefetch

    // --- lane-parallel neighbor resolution: lane g in [0,9) owns (dx,dy) ---
    int nbr0 = -1, nbr1 = -1, nbr2 = -1;   // dz = -1, 0, +1
    if (lane < 9) {
        int dx = lane / 3 - 1;
        int dy = lane % 3 - 1;
        int qm1 = key + (dx * PDIM + dy) * PDIM - 1;   // key at dz = -1
        int lo = 0, hi = n;
        while (lo < hi) {                               // lower_bound(qm1)
            int mid = (lo + hi) >> 1;
            if (skeys[mid] < qm1) lo = mid + 1; else hi = mid;
        }
        int p = lo;
        // check qm1, qm1+1, qm1+2 (sorted unique keys -> p advances <= 3)
        if (p < n) {
            int sk = skeys[p];
            for (int t = 0; t < 3; ++t) {
                int q = qm1 + t;
                while (sk < q && ++p < n) sk = skeys[p];
                if (p >= n) break;
                if (sk == q) {
                    int j = order[p];
                    if (t == 0) nbr0 = j; else if (t == 1) nbr1 = j; else nbr2 = j;
                }
            }
        }
    }

    // --- MAC: 27 taps, neighbor index broadcast from owner lane ---
    int c = lane;              // C == 32; guarded anyway
    float acc = 0.0f;
    #pragma unroll
    for (int k = 0; k < 27; ++k) {
        int g = k / 3;         // (dx,dy) owner lane
        int t = k % 3;         // dz + 1
        int v = (t == 0) ? nbr0 : (t == 1) ? nbr1 : nbr2;
        int j = __shfl(v, g, 32);              // wave32 broadcast (ds_bpermute)
        if (j >= 0) {                           // uniform branch after shfl
            if (c < C) acc += feats[(size_t)j * C + c] * wlds[c * 27 + k];
        }
    }
    if (c < C) out[(size_t)i * C + c] = acc;
}

// ---------------------------------------------------------------------------
static inline size_t alignUp256(size_t x) { return (x + 255) & ~(size_t)255; }

extern "C" void kernel_launch(void* const* d_in, const int* in_sizes, int n_in,
                              void* d_out, int out_size, void* d_ws, size_t ws_size,
                              hipStream_t stream) {
    const float* feats  = (const float*)d_in[0];
    const int*   coords = (const int*)d_in[1];
    const float* weight = (const float*)d_in[2];
    const int N = in_sizes[1] / 4;
    const int C = in_sizes[2] / 27;

    float* out        = (float*)d_out;
    float* out_coords = out + (size_t)N * C;

    // workspace carve-out (~12.3 MB)
    uint8_t* w = (uint8_t*)d_ws;
    int*      keys    = (int*)w;        w += alignUp256((size_t)N * 4);
    unsigned* counts  = (unsigned*)w;   w += alignUp256((size_t)NBUCK * 4);
    unsigned* cursors = (unsigned*)w;   w += alignUp256((size_t)NBUCK * 4);
    unsigned* bstart  = (unsigned*)w;   w += alignUp256((size_t)(NBUCK + 1) * 4);
    int*      tmpKey  = (int*)w;        w += alignUp256((size_t)N * 4);
    int*      tmpIdx  = (int*)w;        w += alignUp256((size_t)N * 4);
    int*      skeys   = (int*)w;        w += alignUp256((size_t)N * 4);
    int*      order   = (int*)w;        w += alignUp256((size_t)N * 4);
    (void)ws_size; (void)n_in; (void)out_size;

    const int gN = (N + TPB - 1) / TPB;
    const int gB = (NBUCK + TPB - 1) / TPB;

    k_zero<<<gB, TPB, 0, stream>>>(counts, NBUCK);
    k_zero<<<gB, TPB, 0, stream>>>(cursors, NBUCK);
    k_keys_hist<<<gN, TPB, 0, stream>>>(coords, keys, counts, N);
    k_scan<<<1, TPB, 0, stream>>>(counts, bstart, NBUCK);
    k_scatter<<<gN, TPB, 0, stream>>>(keys, bstart, cursors, tmpKey, tmpIdx, N);
    k_rank<<<NBUCK, TPB, 0, stream>>>(tmpKey, tmpIdx, bstart, skeys, order);

    const long long threads = (long long)N * 32;
    const int gC = (int)((threads + TPB - 1) / TPB);
    k_conv<<<gC, TPB, 0, stream>>>(feats, coords, weight, skeys, order,
                                   out, out_coords, N, C);
}